// TensorProductConvLayer_79834852098713
// MI455X (gfx1250) — compile-verified
//
#include <hip/hip_runtime.h>

typedef __attribute__((ext_vector_type(2))) float v2f;
typedef __attribute__((ext_vector_type(8))) float v8f;

// ---------------- LDS layout (float offsets), ~250 KB dynamic shared ----------------
#define W1_O   0        // 64 x 80   (W1, row-padded 64->80 for bank spread)
#define B1_O   5120     // 64
#define W2_O   5184     // 64 x 400  (W2 natural layout; stride 400 is conflict-free)
#define B2_O   30784    // 400
#define XA_O   31184    // 64 x 68   (edge_attr tile, reused as h after GEMM1)
#define DN_O   35536    // 64 x 404  (dense = h@W2+b2 for the 64-edge block)
#define XS_O   61392    // 64 x 40   (per-edge staging: x[28], sh[4], dot_vv[4])
#define LDS_FLOATS 63952

#define CS2 0.17677669529f   // (1/sqrt(16)) * (1/sqrt(2))
#define CV2 0.35355339059f   // (1/sqrt(4))  * (1/sqrt(2))
#define C3  0.57735026919f   // 1/sqrt(3)

__global__ void __launch_bounds__(256) init_ws_kernel(float* ws, int n) {
    int t = blockIdx.x * 256 + threadIdx.x;
    if (t < n) ws[t] = 0.0f;
}

__global__ void __launch_bounds__(256) tpconv_main_kernel(
    const float* __restrict__ node_attr, const float* __restrict__ edge_attr,
    const float* __restrict__ edge_sh,   const int*   __restrict__ edge_index,
    const float* __restrict__ W1, const float* __restrict__ b1,
    const float* __restrict__ W2, const float* __restrict__ b2,
    float* __restrict__ sum_ws, float* __restrict__ cnt_ws,
    int E, int nblocks)
{
    extern __shared__ float sh[];
    const int tid  = threadIdx.x;
    const int lane = tid & 31;          // wave32
    const int w    = tid >> 5;          // wave id 0..7
    const int half = lane >> 4;         // lane half (0/1)
    const int ln   = lane & 15;

    // ---- load weights into LDS once per (persistent) workgroup ----
    for (int i = tid; i < 4096;  i += 256) sh[W1_O + (i >> 6) * 80 + (i & 63)] = W1[i];
    for (int i = tid; i < 25600; i += 256) sh[W2_O + i] = W2[i];
    for (int i = tid; i < 64;    i += 256) sh[B1_O + i] = b1[i];
    for (int i = tid; i < 400;   i += 256) sh[B2_O + i] = b2[i];
    __syncthreads();

    for (int eb = blockIdx.x; eb < nblocks; eb += gridDim.x) {
        const int ebase = eb * 64;

        // ---- stage edge_attr tile (64 edges x 64 feats), zero-pad the tail ----
        for (int i = tid; i < 1024; i += 256) {
            const int e  = i >> 4;
            const int k4 = (i & 15) << 2;
            const int ge = ebase + e;
            float4 v = make_float4(0.f, 0.f, 0.f, 0.f);
            if (ge < E) v = *(const float4*)(edge_attr + (size_t)ge * 64 + k4);
            *(float4*)&sh[XA_O + e * 68 + k4] = v;
        }
        __syncthreads();

        // ---- GEMM1: h = relu(Xa @ W1 + b1) via fp32 WMMA 16x16x4 ----
        // wave w: N-tile (w&3), M-subtiles {2*(w>>2), 2*(w>>2)+1}
        {
            const int n0  = (w & 3) * 16;
            const int msb = (w >> 2) * 2;
            v8f acc0 = {}; v8f acc1 = {};
            #pragma unroll
            for (int kk = 0; kk < 16; ++kk) {
                const int k0 = kk * 4 + 2 * half;
                v2f bm;
                bm[0] = sh[W1_O + (k0    ) * 80 + n0 + ln];
                bm[1] = sh[W1_O + (k0 + 1) * 80 + n0 + ln];
                v2f a0 = *(const v2f*)&sh[XA_O + ((msb    ) * 16 + ln) * 68 + k0];
                v2f a1 = *(const v2f*)&sh[XA_O + ((msb + 1) * 16 + ln) * 68 + k0];
                acc0 = __builtin_amdgcn_wmma_f32_16x16x4_f32(false, a0, false, bm, (short)0, acc0, false, false);
                acc1 = __builtin_amdgcn_wmma_f32_16x16x4_f32(false, a1, false, bm, (short)0, acc1, false, false);
            }
            __syncthreads();   // everyone done reading Xa before we overwrite with h
            const float bias = sh[B1_O + n0 + ln];
            #pragma unroll
            for (int r = 0; r < 8; ++r) {
                float h0 = acc0[r] + bias; h0 = h0 > 0.f ? h0 : 0.f;
                float h1 = acc1[r] + bias; h1 = h1 > 0.f ? h1 : 0.f;
                sh[XA_O + ((msb    ) * 16 + half * 8 + r) * 68 + n0 + ln] = h0;
                sh[XA_O + ((msb + 1) * 16 + half * 8 + r) * 68 + n0 + ln] = h1;
            }
        }
        __syncthreads();

        // ---- GEMM2: dense = h @ W2 + b2 ----
        // Preload the full A operand (h, 64x64) into 128 VGPRs, reused across all
        // of this wave's N-tiles; inner loop is then 2 ds_load_b32 + 4 WMMA per k-step.
        {
            v2f ah[64];
            #pragma unroll
            for (int ms = 0; ms < 4; ++ms) {
                #pragma unroll
                for (int kk = 0; kk < 16; ++kk) {
                    ah[ms * 16 + kk] =
                        *(const v2f*)&sh[XA_O + (ms * 16 + ln) * 68 + kk * 4 + 2 * half];
                }
            }
            for (int nt = w; nt < 25; nt += 8) {
                const int n0 = nt * 16;
                v8f c0 = {}; v8f c1 = {}; v8f c2 = {}; v8f c3 = {};
                #pragma unroll
                for (int kk = 0; kk < 16; ++kk) {
                    const int k0 = kk * 4 + 2 * half;
                    v2f bm;
                    bm[0] = sh[W2_O + (k0    ) * 400 + n0 + ln];
                    bm[1] = sh[W2_O + (k0 + 1) * 400 + n0 + ln];
                    c0 = __builtin_amdgcn_wmma_f32_16x16x4_f32(false, ah[     kk], false, bm, (short)0, c0, false, false);
                    c1 = __builtin_amdgcn_wmma_f32_16x16x4_f32(false, ah[16 + kk], false, bm, (short)0, c1, false, false);
                    c2 = __builtin_amdgcn_wmma_f32_16x16x4_f32(false, ah[32 + kk], false, bm, (short)0, c2, false, false);
                    c3 = __builtin_amdgcn_wmma_f32_16x16x4_f32(false, ah[48 + kk], false, bm, (short)0, c3, false, false);
                }
                const float bias = sh[B2_O + n0 + ln];
                #pragma unroll
                for (int r = 0; r < 8; ++r) {
                    sh[DN_O + (     half * 8 + r) * 404 + n0 + ln] = c0[r] + bias;
                    sh[DN_O + (16 + half * 8 + r) * 404 + n0 + ln] = c1[r] + bias;
                    sh[DN_O + (32 + half * 8 + r) * 404 + n0 + ln] = c2[r] + bias;
                    sh[DN_O + (48 + half * 8 + r) * 404 + n0 + ln] = c3[r] + bias;
                }
            }
        }
        __syncthreads();

        // ---- stage per-edge x / sh / dot_vv ----
        if (tid < 64) {
            const int ge = ebase + tid;
            float* xs = &sh[XS_O + tid * 40];
            if (ge < E) {
                const int dst = edge_index[E + ge];
                #pragma unroll
                for (int j = 0; j < 28; ++j) xs[j] = node_attr[(size_t)dst * 28 + j];
                #pragma unroll
                for (int j = 0; j < 4; ++j)  xs[28 + j] = edge_sh[(size_t)ge * 4 + j];
                #pragma unroll
                for (int v = 0; v < 4; ++v) {
                    float d = xs[16 + 3 * v] * xs[29] + xs[17 + 3 * v] * xs[30] + xs[18 + 3 * v] * xs[31];
                    xs[32 + v] = d * C3;
                }
            }
        }
        __syncthreads();

        // ---- tensor product + scatter: 28 components x 64 edges over 256 lanes ----
        for (int t = tid; t < 2048; t += 256) {
            const int e  = t >> 5;
            const int c  = t & 31;
            const int ge = ebase + e;
            if (c >= 28 || ge >= E) continue;
            const float* xs = &sh[XS_O + e * 40];
            const float* dn = &sh[DN_O + e * 404];
            const int src   = edge_index[ge];
            const float shs = xs[28];
            float out;
            if (c < 16) {
                float s1 = 0.f, s2 = 0.f;
                #pragma unroll
                for (int i = 0; i < 16; ++i) s1 += xs[i] * dn[i * 16 + c];            // w_ss
                #pragma unroll
                for (int v = 0; v < 4;  ++v) s2 += xs[32 + v] * dn[256 + v * 16 + c]; // w_vs
                out = s1 * shs * CS2 + s2 * CV2;
                if (c == 0) atomicAdd(&cnt_ws[src], 1.0f);
            } else {
                const int oc = c - 16;
                const int o  = oc / 3;
                const int d  = oc - 3 * o;
                float t1 = 0.f, t2 = 0.f;
                #pragma unroll
                for (int i = 0; i < 16; ++i) t1 += xs[i] * dn[320 + i * 4 + o];              // w_sv
                #pragma unroll
                for (int v = 0; v < 4;  ++v) t2 += xs[16 + 3 * v + d] * dn[384 + v * 4 + o]; // w_vv
                out = t1 * xs[29 + d] * CS2 + t2 * shs * CV2;
            }
            atomicAdd(&sum_ws[(size_t)src * 28 + c], out);
        }
        __syncthreads();   // protect LDS before next block iteration
    }
}

__global__ void __launch_bounds__(256) finalize_kernel(
    const float* __restrict__ sum_ws, const float* __restrict__ cnt_ws,
    const float* __restrict__ node_attr, float* __restrict__ out, int N)
{
    int t = blockIdx.x * 256 + threadIdx.x;
    int n = t >> 5, c = t & 31;
    if (n >= N || c >= 28) return;
    float cnt = cnt_ws[n];
    cnt = cnt > 1.f ? cnt : 1.f;
    out[(size_t)n * 28 + c] = sum_ws[(size_t)n * 28 + c] / cnt + node_attr[(size_t)n * 28 + c];
}

extern "C" void kernel_launch(void* const* d_in, const int* in_sizes, int n_in,
                              void* d_out, int out_size, void* d_ws, size_t ws_size,
                              hipStream_t stream) {
    (void)n_in; (void)out_size; (void)ws_size;
    const float* node_attr  = (const float*)d_in[0];
    const float* edge_attr  = (const float*)d_in[1];
    const float* edge_sh    = (const float*)d_in[2];
    const int*   edge_index = (const int*)  d_in[3];
    const float* W1 = (const float*)d_in[4];
    const float* b1 = (const float*)d_in[5];
    const float* W2 = (const float*)d_in[6];
    const float* b2 = (const float*)d_in[7];

    const int E = in_sizes[1] / 64;
    const int N = in_sizes[0] / 28;

    float* sum_ws = (float*)d_ws;              // N*28 floats
    float* cnt_ws = sum_ws + (size_t)N * 28;   // N floats

    const int zn = N * 29;
    hipLaunchKernelGGL(init_ws_kernel, dim3((zn + 255) / 256), dim3(256), 0, stream,
                       sum_ws, zn);

    const int nblocks = (E + 63) / 64;
    const int grid = nblocks < 2048 ? nblocks : 2048;
    const size_t lds_bytes = (size_t)LDS_FLOATS * sizeof(float);
    hipLaunchKernelGGL(tpconv_main_kernel, dim3(grid), dim3(256), lds_bytes, stream,
                       node_attr, edge_attr, edge_sh, edge_index, W1, b1, W2, b2,
                       sum_ws, cnt_ws, E, nblocks);

    const int ft = N * 32;
    hipLaunchKernelGGL(finalize_kernel, dim3((ft + 255) / 256), dim3(256), 0, stream,
                       sum_ws, cnt_ws, node_attr, (float*)d_out, N);
}